// AvgPoolingModel_22265110462945
// MI455X (gfx1250) — compile-verified
//
#include <hip/hip_runtime.h>
#include <hip/hip_bf16.h>

// ---- problem constants (match reference) ----
#define VOCAB 100000
#define EMB   300
#define BATCH 4096
#define SEQL  200

typedef __attribute__((ext_vector_type(2))) float v2f;
typedef __attribute__((ext_vector_type(4))) float v4f;
typedef __attribute__((ext_vector_type(8))) float v8f;

// =====================================================================
// Kernel 1: embedding gather + mean pool (the bandwidth-bound kernel).
// One 64-thread block per batch row; float4 (b128) loads: 75 float4 per
// 1200B row -> thread t owns f4 slot t, threads 0..10 also own slot 64+t.
// Indices are block-uniform -> scalar loads. Table (120MB) is resident
// in the 192MB L2, so this runs at L2 bandwidth; prefetch hides L2->L0.
// =====================================================================
__global__ __launch_bounds__(64) void pool_kernel(
    const int* __restrict__ batch, const int* __restrict__ lens,
    const float* __restrict__ emb, float* __restrict__ pooled) {
  const int b   = blockIdx.x;
  const int tid = threadIdx.x;                    // 0..63
  const int* __restrict__ row_idx = batch + (size_t)b * SEQL;
  const v4f* __restrict__ emb4 = (const v4f*)emb; // rows are 16B aligned

  v4f acc0 = {0.f, 0.f, 0.f, 0.f};
  v4f acc1 = {0.f, 0.f, 0.f, 0.f};

  for (int t = 0; t < SEQL; ++t) {
    const int idx = row_idx[t];
    const v4f* __restrict__ r = emb4 + (size_t)idx * (EMB / 4);

    // Prefetch the row needed 8 tokens ahead (10 x 128B covers 1200B).
    if (t + 8 < SEQL && tid < 10) {
      const int idx2 = row_idx[t + 8];
      const char* p = (const char*)(emb + (size_t)idx2 * EMB) + tid * 128;
      __builtin_prefetch(p, 0, 0);
    }

    acc0 += r[tid];
    if (tid < (EMB / 4) - 64) acc1 += r[64 + tid];
  }

  const float inv = 1.0f / (float)lens[b];
  v4f* __restrict__ o = (v4f*)(pooled + (size_t)b * EMB);
  o[tid] = acc0 * inv;
  if (tid < (EMB / 4) - 64) o[64 + tid] = acc1 * inv;
}

// =====================================================================
// Kernel 2: fused MLP via V_WMMA_F32_16X16X4_F32 (wave32).
// One block (128 thr = 4 waves) per 16-row batch tile.
//   GEMM1: [16x300] x [300x160pad]  (75 WMMAs per N-tile)
//   GEMM2: [16x152pad] x [152x160pad]
//   GEMM3: [16x150] . W3 -> 8-lane shuffle reduction
// A layout : lane&15 = M, K = k0 + 2*(lane>>4) (+1 for .y)
// B layout : lane&15 = N, same K striping; contiguous-k in row-major W
// C layout : VGPR v -> M = v + 8*(lane>=16), N = lane&15
// Out-of-range N lanes read clamped (finite) weight rows; their garbage
// accumulators are zeroed only at store time -> no masking in the loop.
// =====================================================================
#define SPW 305   // pooled tile LDS stride (odd -> conflict-free)
#define SHW 161   // hidden tile LDS stride (odd)

__global__ __launch_bounds__(128) void mlp_wmma_kernel(
    const float* __restrict__ pooled,
    const float* __restrict__ W1, const float* __restrict__ b1,
    const float* __restrict__ W2, const float* __restrict__ b2,
    const float* __restrict__ W3, const float* __restrict__ b3,
    float* __restrict__ out) {
  __shared__ float sP [16][SPW];
  __shared__ float sH1[16][SHW];
  __shared__ float sH2[16][SHW];

  const int tid = threadIdx.x;
  const int m0  = blockIdx.x * 16;

  // Stage pooled tile (zero-pad K beyond 300).
  for (int i = tid; i < 16 * SPW; i += 128) {
    const int m = i / SPW, k = i - m * SPW;
    float v = 0.0f;
    if (k < EMB) v = pooled[(size_t)(m0 + m) * EMB + k];
    sP[m][k] = v;
  }
  __syncthreads();

  const int lane = tid & 31;
  const int wave = tid >> 5;
  const int ln   = lane & 15;
  const int koff = (lane >> 4) << 1;          // 0 or 2
  const int mrow = (lane < 16) ? 0 : 8;

  // ---------------- layer 1: h1 = relu(pooled @ W1^T + b1) -------------
  for (int nt = wave; nt < 10; nt += 4) {     // N tiles cover 0..159
    const int   n_g   = nt * 16 + ln;
    const int   n_c   = (n_g < 150) ? n_g : 149;
    const float nmask = (n_g < 150) ? 1.0f : 0.0f;
    const float* __restrict__ Bp = W1 + (size_t)n_c * EMB;

    v8f c = {};
    for (int k0 = 0; k0 < EMB; k0 += 4) {
      v2f a, bv;
      a.x  = sP[ln][k0 + koff];
      a.y  = sP[ln][k0 + koff + 1];
      bv.x = Bp[k0 + koff];
      bv.y = Bp[k0 + koff + 1];
      c = __builtin_amdgcn_wmma_f32_16x16x4_f32(false, a, false, bv,
                                                (short)0, c, false, false);
    }
    const float bias = b1[n_c];
#pragma unroll
    for (int v = 0; v < 8; ++v) {
      float h = c[v] + bias;
      h = (h > 0.0f) ? h : 0.0f;
      sH1[mrow + v][n_g] = h * nmask;         // cols 150..159 forced to 0
    }
  }
  __syncthreads();

  // ---------------- layer 2: h2 = relu(h1 @ W2^T + b2) -----------------
  for (int nt = wave; nt < 10; nt += 4) {
    const int   n_g   = nt * 16 + ln;
    const int   n_c   = (n_g < 150) ? n_g : 149;
    const float nmask = (n_g < 150) ? 1.0f : 0.0f;
    const float* __restrict__ Bp = W2 + (size_t)n_c * 150;

    v8f c = {};
    for (int k0 = 0; k0 < 152; k0 += 4) {     // K padded 150 -> 152
      const int k  = k0 + koff;
      const int kc = (k <= 148) ? k : 148;    // finite addr; a==0 there
      v2f a, bv;
      a.x  = sH1[ln][k];
      a.y  = sH1[ln][k + 1];
      bv.x = Bp[kc];
      bv.y = Bp[kc + 1];
      c = __builtin_amdgcn_wmma_f32_16x16x4_f32(false, a, false, bv,
                                                (short)0, c, false, false);
    }
    const float bias = b2[n_c];
#pragma unroll
    for (int v = 0; v < 8; ++v) {
      float h = c[v] + bias;
      h = (h > 0.0f) ? h : 0.0f;
      sH2[mrow + v][n_g] = h * nmask;
    }
  }
  __syncthreads();

  // ---------------- layer 3: out = h2 . W3 + b3 ------------------------
  const int m = tid >> 3;        // 16 rows x 8 lanes each
  const int s = tid & 7;
  float partial = 0.0f;
  for (int k = s; k < 150; k += 8) partial += sH2[m][k] * W3[k];
  partial += __shfl_xor(partial, 1);
  partial += __shfl_xor(partial, 2);
  partial += __shfl_xor(partial, 4);
  if (s == 0) out[m0 + m] = partial + b3[0];
}

// =====================================================================
extern "C" void kernel_launch(void* const* d_in, const int* in_sizes, int n_in,
                              void* d_out, int out_size, void* d_ws, size_t ws_size,
                              hipStream_t stream) {
  const int*   batch = (const int*)d_in[0];
  const int*   lens  = (const int*)d_in[1];
  const float* emb   = (const float*)d_in[2];
  const float* W1    = (const float*)d_in[3];
  const float* b1    = (const float*)d_in[4];
  const float* W2    = (const float*)d_in[5];
  const float* b2    = (const float*)d_in[6];
  const float* W3    = (const float*)d_in[7];
  const float* b3    = (const float*)d_in[8];
  float*       out   = (float*)d_out;

  float* pooled = (float*)d_ws;   // BATCH*EMB floats = 4.9 MB scratch

  pool_kernel<<<BATCH, 64, 0, stream>>>(batch, lens, emb, pooled);
  mlp_wmma_kernel<<<BATCH / 16, 128, 0, stream>>>(pooled, W1, b1, W2, b2,
                                                  W3, b3, out);
}